// SparseAttention_970662609474
// MI455X (gfx1250) — compile-verified
//
#include <hip/hip_runtime.h>
#include <hip/hip_bf16.h>
#include <math.h>

typedef __attribute__((ext_vector_type(2))) float v2f;
typedef __attribute__((ext_vector_type(4))) float v4f;
typedef __attribute__((ext_vector_type(8))) float v8f;

#define H_Q    32
#define H_KV   8
#define GRP    4
#define DIM    128
#define PSZ    32
#define SINKN  4
#define TOPK   16
#define NPAGE  2048
#define HIDDEN 4096
#define NCOLS  6144              // 4096 q + 1024 k + 1024 v
#define T_TOK  (SINKN + TOPK * PSZ)   // 516
#define T_PAD  528                    // 33 * 16
#define NTILE  (T_PAD / 16)           // 33
#define SCALE  0.08838834764831845f   // 1/sqrt(128)
#define NEGINF (-1e30f)
#define SENT   0xFFFFFFFFu            // decode-token / padding sentinel

// workspace layout in floats
#define WS_QKV   0        // 6144   accumulator for x@[Wq|Wk|Wv]
#define WS_QROT  6144     // 4096   rotated q [32][128]
#define WS_KNEW  10240    // 1024   rotated k [8][128]
#define WS_VNEW  11264    // 1024   v [8][128]
#define WS_QSUM  12288    // 1024   sum of q over GQA group [8][128]
#define WS_LOGIT 13312    // 16384  group logits [8][2048]
#define WS_TOPP  29696    // 128    top pages (int) [8][16]

__device__ __forceinline__ float wave_sum(float v) {
  #pragma unroll
  for (int off = 16; off > 0; off >>= 1) v += __shfl_xor(v, off, 32);
  return v;
}
__device__ __forceinline__ float wave_max(float v) {
  #pragma unroll
  for (int off = 16; off > 0; off >>= 1) v = fmaxf(v, __shfl_xor(v, off, 32));
  return v;
}

// ---------------------------------------------------------------- zero accum
__global__ void zero_qkv_kernel(float* __restrict__ acc) {
  for (int i = threadIdx.x; i < NCOLS; i += 256) acc[i] = 0.0f;
}

// ------------------------------------------------------- fused QKV matvec
// grid (24, 16), block 256.  Each block: 256 cols x 256 rows, atomic reduce.
__global__ void qkv_matvec_kernel(const float* __restrict__ x,
                                  const float* __restrict__ Wq,
                                  const float* __restrict__ Wk,
                                  const float* __restrict__ Wv,
                                  float* __restrict__ acc) {
  __shared__ float xs[256];
  const int rbase = blockIdx.y * 256;
  xs[threadIdx.x] = x[rbase + threadIdx.x];
  __syncthreads();

  const int col = blockIdx.x * 256 + (int)threadIdx.x;
  const float* W;
  int ldw, c;
  if (col < HIDDEN)            { W = Wq; ldw = HIDDEN; c = col; }
  else if (col < HIDDEN + 1024){ W = Wk; ldw = 1024;   c = col - HIDDEN; }
  else                         { W = Wv; ldw = 1024;   c = col - HIDDEN - 1024; }

  const float* wp = W + (size_t)rbase * ldw + c;
  float s = 0.0f;
  #pragma unroll 8
  for (int r = 0; r < 256; ++r) {
    s += xs[r] * __builtin_nontemporal_load(wp);    // one-shot weight stream
    wp += ldw;
  }
  atomicAdd(&acc[col], s);
}

// --------------------------------------------------------------- RoPE etc.
__global__ void rope_kernel(const float* __restrict__ qkv,
                            const float* __restrict__ cosv,
                            const float* __restrict__ sinv,
                            float* __restrict__ qrot,
                            float* __restrict__ knew,
                            float* __restrict__ vnew,
                            float* __restrict__ qsum) {
  const int tid = threadIdx.x;
  for (int i = tid; i < H_Q * DIM; i += 256) {
    int d = i & (DIM - 1);
    int base = i - d;
    float t = qkv[i];
    float rot = (d < 64) ? -qkv[base + d + 64] : qkv[base + d - 64];
    qrot[i] = t * cosv[d] + rot * sinv[d];
  }
  for (int i = tid; i < H_KV * DIM; i += 256) {
    int d = i & (DIM - 1);
    int base = HIDDEN + (i - d);
    float t = qkv[HIDDEN + i];
    float rot = (d < 64) ? -qkv[base + d + 64] : qkv[base + d - 64];
    knew[i] = t * cosv[d] + rot * sinv[d];
  }
  for (int i = tid; i < H_KV * DIM; i += 256) vnew[i] = qkv[HIDDEN + 1024 + i];
  for (int i = tid; i < H_KV * DIM; i += 256) {
    int d = i & (DIM - 1);
    int h = i >> 7;
    float s = 0.0f;
    #pragma unroll
    for (int g = 0; g < GRP; ++g) {
      int idx = (h * GRP + g) * DIM + d;
      int base = idx - d;
      float t = qkv[idx];
      float rot = (d < 64) ? -qkv[base + d + 64] : qkv[base + d - 64];
      s += t * cosv[d] + rot * sinv[d];
    }
    qsum[i] = s;
  }
}

// -------------------------------------------- fused page-pool + Quest logit
// grid 2048, block 256 (8 waves). One wave per (kv head, page).
// Streams all of paged_k (256 MB) once: nontemporal b128 loads, no per-slot
// branching; decode token handled as a one-shot fixup on its page.
__global__ __launch_bounds__(256)
void pool_logits_kernel(const float* __restrict__ paged_k,
                        const float* __restrict__ knew,
                        const float* __restrict__ qsum,
                        const int* __restrict__ clp,
                        float* __restrict__ logits) {
  const int wave = threadIdx.x >> 5;
  const int lane = threadIdx.x & 31;
  const int idx = blockIdx.x * 8 + wave;           // 0..16383
  const int h = idx >> 11;
  const int p = idx & (NPAGE - 1);

  const int cache_len = *clp;
  const int seq = cache_len + 1;
  int vcnt = seq - p * PSZ;
  vcnt = vcnt < 0 ? 0 : (vcnt > PSZ ? PSZ : vcnt);

  const float* rp = paged_k + ((size_t)(h * NPAGE + p) * PSZ) * DIM + lane * 4;
  v4f acc = (v4f)(0.0f);
  if (vcnt == PSZ) {
    #pragma unroll 8
    for (int s = 0; s < PSZ; ++s)
      acc += __builtin_nontemporal_load((const v4f*)(rp + s * DIM));
  } else {
    for (int s = 0; s < vcnt; ++s)
      acc += __builtin_nontemporal_load((const v4f*)(rp + s * DIM));
  }

  // decode-token fixup: replace stale cache row with rotated k_new
  if (p == (cache_len / PSZ)) {
    int off = cache_len & (PSZ - 1);                // off < vcnt by construction
    v4f old = *(const v4f*)(rp + off * DIM);
    v4f kn  = *(const v4f*)(knew + h * DIM + lane * 4);
    acc += kn - old;
  }

  const float inv = 1.0f / (float)(vcnt > 0 ? vcnt : 1);
  v4f qs = *(const v4f*)(qsum + h * DIM + lane * 4);
  float part = (acc.x * qs.x + acc.y * qs.y + acc.z * qs.z + acc.w * qs.w) * inv;
  part = wave_sum(part);
  if (lane == 0) {
    float lg = part * SCALE;
    if (p == 0 || vcnt == 0) lg = NEGINF;
    logits[h * NPAGE + p] = lg;
  }
}

// ----------------------------------------------------------------- top-16
__global__ void topk_kernel(const float* __restrict__ logits,
                            int* __restrict__ top_pages) {
  __shared__ float vals[NPAGE];
  __shared__ float rv[256];
  __shared__ int   ri[256];
  const int h = blockIdx.x;
  const int tid = threadIdx.x;
  for (int i = tid; i < NPAGE; i += 256) vals[i] = logits[h * NPAGE + i];
  __syncthreads();

  for (int sel = 0; sel < TOPK; ++sel) {
    float bv = -3.0e38f; int bi = NPAGE;
    for (int i = tid; i < NPAGE; i += 256)
      if (vals[i] > bv) { bv = vals[i]; bi = i; }   // first (lowest idx) wins ties
    rv[tid] = bv; ri[tid] = bi;
    __syncthreads();
    for (int off = 128; off > 0; off >>= 1) {
      if (tid < off) {
        float v2 = rv[tid + off]; int i2 = ri[tid + off];
        if (v2 > rv[tid] || (v2 == rv[tid] && i2 < ri[tid])) { rv[tid] = v2; ri[tid] = i2; }
      }
      __syncthreads();
    }
    if (tid == 0) { top_pages[h * TOPK + sel] = ri[0]; vals[ri[0]] = -3.0e38f; }
    __syncthreads();
  }
}

// --------------------------------------- sparse sink attention with WMMA f32
// grid H_KV, block 128 (4 waves).  A operands pre-padded in LDS (no EXEC
// games); K/V rows addressed via uint32 offsets so loads stay global_*.
__global__ __launch_bounds__(128)
void attn_kernel(const float* __restrict__ paged_k,
                 const float* __restrict__ paged_v,
                 const float* __restrict__ qrot,
                 const float* __restrict__ knew,
                 const float* __restrict__ vnew,
                 const int* __restrict__ top_pages,
                 const int* __restrict__ clp,
                 float* __restrict__ out) {
  __shared__ float    sh_qA[16][DIM];       // rows >= GRP zeroed (A padding)
  __shared__ float    sh_probs[16][T_PAD];  // scores -> probs; rows >= GRP zero
  __shared__ unsigned sh_toff[T_PAD];       // element offset into paged_k/v, or SENT
  __shared__ int      sh_mask[T_PAD];

  const int h = blockIdx.x;
  const int tid = threadIdx.x;
  const int w = tid >> 5;
  const int lane = tid & 31;
  const int half = lane >> 4;
  const int l16 = lane & 15;

  const int cache_len = *clp;
  const int seq = cache_len + 1;

  // token table
  for (int t = tid; t < T_PAD; t += 128) {
    int page, slot, pos, valid;
    if (t < SINKN)      { page = 0; slot = t; pos = t; valid = (pos < seq); }
    else if (t < T_TOK) {
      int i = (t - SINKN) / PSZ;
      slot = (t - SINKN) % PSZ;
      page = top_pages[h * TOPK + i];
      pos  = page * PSZ + slot;
      valid = (pos < seq);
    } else { page = 0; slot = 0; pos = seq; valid = 0; }

    unsigned off;
    if (!valid || pos == cache_len) off = SENT;     // decode token or padding
    else off = (unsigned)((h * NPAGE + page) * PSZ + slot) * DIM;
    sh_toff[t] = off;
    sh_mask[t] = valid;
  }
  // padded Q
  for (int i = tid; i < 16 * DIM; i += 128) {
    int r = i >> 7, d = i & (DIM - 1);
    sh_qA[r][d] = (r < GRP) ? qrot[(h * GRP + r) * DIM + d] : 0.0f;
  }
  // zero padding rows of probs (rows 0..3 fully written by score phase)
  for (int i = tid; i < (16 - GRP) * T_PAD; i += 128)
    sh_probs[GRP + i / T_PAD][i % T_PAD] = 0.0f;
  __syncthreads();

  // ---- scores = Q K^T * scale  (two interleaved WMMA chains)
  for (int tile = w; tile < NTILE; tile += 4) {
    const int nbase = tile * 16;
    unsigned off = sh_toff[nbase + l16];
    const float* kr = (off == SENT) ? (knew + h * DIM) : (paged_k + (size_t)off);
    v8f acc0 = (v8f)(0.0f), acc1 = (v8f)(0.0f);
    #pragma unroll 2
    for (int kk = 0; kk < DIM / 4; kk += 2) {
      const int d0 = kk * 4 + half * 2;
      v2f a0 = *(const v2f*)&sh_qA[l16][d0];
      v2f a1 = *(const v2f*)&sh_qA[l16][d0 + 4];
      v2f b0 = *(const v2f*)(kr + d0);
      v2f b1 = *(const v2f*)(kr + d0 + 4);
      acc0 = __builtin_amdgcn_wmma_f32_16x16x4_f32(false, a0, false, b0, (short)0,
                                                   acc0, false, false);
      acc1 = __builtin_amdgcn_wmma_f32_16x16x4_f32(false, a1, false, b1, (short)0,
                                                   acc1, false, false);
    }
    v8f acc = acc0 + acc1;
    if (half == 0) {                                 // lanes 0-15 hold rows 0..7
      #pragma unroll
      for (int m = 0; m < GRP; ++m) {
        const int t = nbase + l16;
        float sc = acc[m] * SCALE;
        if (!sh_mask[t]) sc = NEGINF;
        sh_probs[m][t] = sc;
      }
    }
  }
  __syncthreads();

  // ---- softmax per query row (wave w owns row w)
  {
    const int m = w;
    float mx = NEGINF;
    for (int t = lane; t < T_TOK; t += 32) mx = fmaxf(mx, sh_probs[m][t]);
    mx = wave_max(mx);
    float sum = 0.0f;
    for (int t = lane; t < T_PAD; t += 32) {
      float e = __expf(sh_probs[m][t] - mx);        // masked -> underflows to 0
      sh_probs[m][t] = e;
      sum += e;
    }
    sum = wave_sum(sum);
    const float inv = 1.0f / sum;
    for (int t = lane; t < T_PAD; t += 32) sh_probs[m][t] *= inv;
  }
  __syncthreads();

  // ---- out = P V   (two interleaved WMMA chains, K = 528)
  for (int nt = w; nt < DIM / 16; nt += 4) {
    const int nbase = nt * 16;
    v8f acc0 = (v8f)(0.0f), acc1 = (v8f)(0.0f);
    for (int kk = 0; kk < T_PAD / 4; kk += 2) {
      const int k0 = kk * 4 + half * 2;             // token index, chain 0
      const int k1 = k0 + 4;                        // token index, chain 1
      v2f a0 = *(const v2f*)&sh_probs[l16][k0];
      v2f a1 = *(const v2f*)&sh_probs[l16][k1];
      unsigned o00 = sh_toff[k0], o01 = sh_toff[k0 + 1];
      unsigned o10 = sh_toff[k1], o11 = sh_toff[k1 + 1];
      const float* r00 = (o00 == SENT) ? (vnew + h * DIM) : (paged_v + (size_t)o00);
      const float* r01 = (o01 == SENT) ? (vnew + h * DIM) : (paged_v + (size_t)o01);
      const float* r10 = (o10 == SENT) ? (vnew + h * DIM) : (paged_v + (size_t)o10);
      const float* r11 = (o11 == SENT) ? (vnew + h * DIM) : (paged_v + (size_t)o11);
      v2f b0, b1;
      b0.x = r00[nbase + l16];
      b0.y = r01[nbase + l16];
      b1.x = r10[nbase + l16];
      b1.y = r11[nbase + l16];
      acc0 = __builtin_amdgcn_wmma_f32_16x16x4_f32(false, a0, false, b0, (short)0,
                                                   acc0, false, false);
      acc1 = __builtin_amdgcn_wmma_f32_16x16x4_f32(false, a1, false, b1, (short)0,
                                                   acc1, false, false);
    }
    v8f acc = acc0 + acc1;
    if (half == 0) {
      #pragma unroll
      for (int m = 0; m < GRP; ++m)
        out[(h * GRP + m) * DIM + nbase + l16] = acc[m];
    }
  }
}

// ----------------------------------------------------------------- launcher
extern "C" void kernel_launch(void* const* d_in, const int* in_sizes, int n_in,
                              void* d_out, int out_size, void* d_ws, size_t ws_size,
                              hipStream_t stream) {
  const float* x    = (const float*)d_in[0];
  const float* cosv = (const float*)d_in[1];
  const float* sinv = (const float*)d_in[2];
  const float* Wq   = (const float*)d_in[3];
  const float* Wk   = (const float*)d_in[4];
  const float* Wv   = (const float*)d_in[5];
  const float* pk   = (const float*)d_in[6];
  const float* pv   = (const float*)d_in[7];
  const int*   clp  = (const int*)d_in[8];

  float* ws  = (float*)d_ws;
  float* out = (float*)d_out;

  zero_qkv_kernel<<<1, 256, 0, stream>>>(ws + WS_QKV);
  qkv_matvec_kernel<<<dim3(NCOLS / 256, HIDDEN / 256), 256, 0, stream>>>(
      x, Wq, Wk, Wv, ws + WS_QKV);
  rope_kernel<<<1, 256, 0, stream>>>(ws + WS_QKV, cosv, sinv,
                                     ws + WS_QROT, ws + WS_KNEW, ws + WS_VNEW,
                                     ws + WS_QSUM);
  pool_logits_kernel<<<(H_KV * NPAGE) / 8, 256, 0, stream>>>(
      pk, ws + WS_KNEW, ws + WS_QSUM, clp, ws + WS_LOGIT);
  topk_kernel<<<H_KV, 256, 0, stream>>>(ws + WS_LOGIT, (int*)(ws + WS_TOPP));
  attn_kernel<<<H_KV, 128, 0, stream>>>(pk, pv, ws + WS_QROT, ws + WS_KNEW,
                                        ws + WS_VNEW, (const int*)(ws + WS_TOPP),
                                        clp, out);
}